// GraphConvolution_38946763440896
// MI455X (gfx1250) — compile-verified
//
#include <hip/hip_runtime.h>

typedef __attribute__((ext_vector_type(2))) float v2f;
typedef __attribute__((ext_vector_type(8))) float v8f;

#define N_NODES 100000
#define N_EDGES 3200000
#define IN_DIM  256
#define OUT_DIM 128

// ---------------------------------------------------------------------------
// Kernel 1: xw[N,128] = x[N,256] @ W[256,128]  (f32 WMMA 16x16x4)
// Block = 256 threads = 8 wave32s. Each wave computes a 16(M) x 128(N) strip,
// so one block covers 128 rows. W is staged through LDS in 64-row chunks.
//
// A (16x4 f32) fragment layout (ISA 7.12.2): lanes 0-15 = rows M=0..15 holding
// K=0 (v0), K=1 (v1); lanes 16-31 = same rows holding K=2,K=3. -> per lane a
// contiguous float2 load at column k + 2*(lane>>4).
// B (4x16 f32): v0 = row K=0 (lanes0-15) / K=2 (lanes16-31); v1 = K=1 / K=3.
// C/D (16x16 f32): vgpr i -> M=i (lanes0-15), M=i+8 (lanes16-31), N=lane%16.
// ---------------------------------------------------------------------------
__global__ __launch_bounds__(256)
void gemm_xw(const float* __restrict__ x, const float* __restrict__ w,
             float* __restrict__ xw) {
  __shared__ float wlds[64 * OUT_DIM];          // 32 KB chunk of W

  const int tid  = threadIdx.x;
  const int wave = tid >> 5;
  const int lane = tid & 31;
  const int lmod = lane & 15;
  const int khalf = (lane >> 4) << 1;           // 0 or 2

  const int m0  = (blockIdx.x * 8 + wave) * 16; // first row of this wave's tile
  int row = m0 + lmod;
  int rowc = row < N_NODES ? row : (N_NODES - 1); // clamp (stores are guarded)
  const float* xrow = x + (size_t)rowc * IN_DIM;

  v8f acc[8];
#pragma unroll
  for (int t = 0; t < 8; ++t) acc[t] = (v8f){0.f,0.f,0.f,0.f,0.f,0.f,0.f,0.f};

  const float4* wsrc = (const float4*)w;
  float4* wdst = (float4*)wlds;

  for (int kc = 0; kc < IN_DIM; kc += 64) {
    __syncthreads();                            // protect previous chunk use
    // stage rows [kc, kc+64) of W: 64*128 floats = 2048 float4 / 256 threads
#pragma unroll
    for (int i = 0; i < 8; ++i)
      wdst[tid + i * 256] = wsrc[kc * (OUT_DIM / 4) + tid + i * 256];
    __syncthreads();

    for (int k = 0; k < 64; k += 4) {
      // A fragment: float2 at (row, kc+k+khalf)
      v2f a = *(const v2f*)(xrow + kc + k + khalf);
#pragma unroll
      for (int t = 0; t < 8; ++t) {
        const int col = t * 16 + lmod;
        v2f b;
        b.x = wlds[(k + khalf) * OUT_DIM + col];
        b.y = wlds[(k + khalf + 1) * OUT_DIM + col];
        acc[t] = __builtin_amdgcn_wmma_f32_16x16x4_f32(
            /*neg_a=*/false, a, /*neg_b=*/false, b,
            /*c_mod=*/(short)0, acc[t], /*reuse_a=*/false, /*reuse_b=*/false);
      }
    }
  }

  // Store D tiles
#pragma unroll
  for (int t = 0; t < 8; ++t) {
    const int col = t * 16 + lmod;
#pragma unroll
    for (int i = 0; i < 8; ++i) {
      const int m = m0 + i + ((lane >> 4) << 3);
      if (m < N_NODES) xw[(size_t)m * OUT_DIM + col] = acc[t][i];
    }
  }
}

// ---------------------------------------------------------------------------
// Kernel 2: out[n][j] = bias[j]  (accumulator seed; atomics add on top)
// ---------------------------------------------------------------------------
__global__ __launch_bounds__(256)
void init_out(float* __restrict__ out, const float* __restrict__ bias) {
  size_t i = (size_t)blockIdx.x * 256 + threadIdx.x;
  if (i < (size_t)N_NODES * OUT_DIM) out[i] = bias[i & (OUT_DIM - 1)];
}

// ---------------------------------------------------------------------------
// Kernel 3: SpMM. One wave32 per edge; lane owns a float4 column chunk
// (32 lanes * 4 = 128 cols). Gather is one coalesced 512B read from xw
// (L2-resident, 51 MB << 192 MB L2); scatter is 4 global_atomic_add_f32.
// ---------------------------------------------------------------------------
__global__ __launch_bounds__(256)
void spmm_edges(const float* __restrict__ xw,
                const int* __restrict__ edge_row,
                const int* __restrict__ edge_col,
                const float* __restrict__ edge_val,
                float* __restrict__ out) {
  const int gtid = blockIdx.x * 256 + threadIdx.x;
  const int e    = gtid >> 5;                   // one wave per edge
  const int lane = gtid & 31;
  if (e >= N_EDGES) return;

  const int   r = edge_row[e];                  // uniform per wave -> scalarized
  const int   c = edge_col[e];
  const float v = edge_val[e];

  const float4 g = *(const float4*)(xw + (size_t)c * OUT_DIM + lane * 4);
  float* o = out + (size_t)r * OUT_DIM + lane * 4;
  atomicAdd(o + 0, v * g.x);
  atomicAdd(o + 1, v * g.y);
  atomicAdd(o + 2, v * g.z);
  atomicAdd(o + 3, v * g.w);
}

// ---------------------------------------------------------------------------
// Kernel 4: in-place ReLU (bias already seeded in init_out)
// ---------------------------------------------------------------------------
__global__ __launch_bounds__(256)
void relu_inplace(float* __restrict__ out) {
  size_t i = (size_t)blockIdx.x * 256 + threadIdx.x;
  if (i < (size_t)N_NODES * OUT_DIM) out[i] = fmaxf(out[i], 0.f);
}

// ---------------------------------------------------------------------------
extern "C" void kernel_launch(void* const* d_in, const int* in_sizes, int n_in,
                              void* d_out, int out_size, void* d_ws, size_t ws_size,
                              hipStream_t stream) {
  const float* x        = (const float*)d_in[0];
  const int*   edge_row = (const int*)  d_in[1];
  const int*   edge_col = (const int*)  d_in[2];
  const float* edge_val = (const float*)d_in[3];
  const float* weight   = (const float*)d_in[4];
  const float* bias     = (const float*)d_in[5];
  float* out = (float*)d_out;
  float* xw  = (float*)d_ws;                    // 100000*128*4 = 51.2 MB scratch

  const int gemm_blocks = (N_NODES + 127) / 128;
  gemm_xw<<<gemm_blocks, 256, 0, stream>>>(x, weight, xw);

  const long long n_out = (long long)N_NODES * OUT_DIM;
  init_out<<<(unsigned)((n_out + 255) / 256), 256, 0, stream>>>(out, bias);

  const long long spmm_threads = (long long)N_EDGES * 32;
  spmm_edges<<<(unsigned)((spmm_threads + 255) / 256), 256, 0, stream>>>(
      xw, edge_row, edge_col, edge_val, out);

  relu_inplace<<<(unsigned)((n_out + 255) / 256), 256, 0, stream>>>(out);
}